// SAGE_56169582297586
// MI455X (gfx1250) — compile-verified
//
#include <hip/hip_runtime.h>

typedef float v2f __attribute__((ext_vector_type(2)));
typedef float v8f __attribute__((ext_vector_type(8)));

#define N_NODES 100000
#define N_EDGES 1600000
#define IN_C    128
#define HID_C   128
#define OUT_C   64

// ---------------- utility: zero a float buffer ----------------
__global__ void sage_zero_f32(float* __restrict__ p, long n) {
  long i = (long)blockIdx.x * blockDim.x + threadIdx.x;
  long stride = (long)gridDim.x * blockDim.x;
  for (; i < n; i += stride) p[i] = 0.0f;
}

// ---------------- degree: atomicAdd 1.0 per edge --------------
__global__ void sage_degree(const int* __restrict__ dst, float* __restrict__ deg, int nE) {
  int e = blockIdx.x * blockDim.x + threadIdx.x;
  if (e < nE) atomicAdd(&deg[dst[e]], 1.0f);
}

// ------- scatter-add: one wave per edge, 128 features ---------
// lane l handles features [4l, 4l+4): float4 gather + 4 f32 atomics.
__global__ void sage_scatter(const float* __restrict__ feat,  // [N,128]
                             const int* __restrict__ src,
                             const int* __restrict__ dst,
                             float* __restrict__ agg,          // [N,128]
                             int nE) {
  long gid = (long)blockIdx.x * blockDim.x + threadIdx.x;
  long total = (long)nE * 32;
  long stride = (long)gridDim.x * blockDim.x;
  for (; gid < total; gid += stride) {
    int e  = (int)(gid >> 5);
    int c4 = (int)(gid & 31) << 2;
    int s = src[e];
    int d = dst[e];
    const float4 v = *(const float4*)(feat + (long)s * 128 + c4);
    float* o = agg + (long)d * 128 + c4;
    atomicAdd(o + 0, v.x);
    atomicAdd(o + 1, v.y);
    atomicAdd(o + 2, v.z);
    atomicAdd(o + 3, v.w);
  }
}

// ---- fused SAGE layer: out = act( (agg/deg)@Wl + X@Wr + b ) ----
// Block = (NCOLS/16) waves * 32 threads. Block b owns rows [16b,16b+16).
// Wave w owns output columns [16w, 16w+16). K = 128, stepped by 4 with
// V_WMMA_F32_16X16X4_F32. Fragment layouts per CDNA5 ISA 7.12.2.
template <int NCOLS, bool RELU>
__global__ void sage_gemm(const float* __restrict__ agg,  // [N,128]
                          const float* __restrict__ deg,  // [N]
                          const float* __restrict__ xin,  // [N,128]
                          const float* __restrict__ Wl,   // [128,NCOLS]
                          const float* __restrict__ Wr,   // [128,NCOLS]
                          const float* __restrict__ bias, // [NCOLS]
                          float* __restrict__ out) {      // [N,NCOLS]
  constexpr int K = 128;
  const int wave = threadIdx.x >> 5;
  const int lane = threadIdx.x & 31;
  const int half = lane >> 4;   // 0: lanes 0-15, 1: lanes 16-31
  const int l16  = lane & 15;
  const int row0 = blockIdx.x * 16;
  const int col0 = wave * 16;

  // per-lane row scale for the mean aggregation (A rows = l16)
  const float dscale = 1.0f / fmaxf(deg[row0 + l16], 1.0f);

  const float* Arow = agg + (long)(row0 + l16) * K;
  const float* Xrow = xin + (long)(row0 + l16) * K;

  v8f c = {};

  // pass 1: (agg/deg) @ Wl
  #pragma unroll 4
  for (int k = 0; k < K; k += 4) {
    v2f a, b;
    a.x = Arow[k + 2 * half]     * dscale;      // A[M=l16][K = k+2h]
    a.y = Arow[k + 2 * half + 1] * dscale;      // A[M=l16][K = k+2h+1]
    b.x = Wl[(k + 2 * half)     * NCOLS + col0 + l16];  // B[K][N=l16]
    b.y = Wl[(k + 2 * half + 1) * NCOLS + col0 + l16];
    c = __builtin_amdgcn_wmma_f32_16x16x4_f32(false, a, false, b,
                                              (short)0, c, false, false);
  }
  // pass 2: X @ Wr (accumulate into same C)
  #pragma unroll 4
  for (int k = 0; k < K; k += 4) {
    v2f a, b;
    a.x = Xrow[k + 2 * half];
    a.y = Xrow[k + 2 * half + 1];
    b.x = Wr[(k + 2 * half)     * NCOLS + col0 + l16];
    b.y = Wr[(k + 2 * half + 1) * NCOLS + col0 + l16];
    c = __builtin_amdgcn_wmma_f32_16x16x4_f32(false, a, false, b,
                                              (short)0, c, false, false);
  }

  const float bval = bias[col0 + l16];
  // C/D layout: VGPR v -> M = v + 8*half, N = l16
  #pragma unroll
  for (int v = 0; v < 8; ++v) {
    const int m = v + 8 * half;
    float val = c[v] + bval;
    if (RELU) val = fmaxf(val, 0.0f);
    out[(long)(row0 + m) * NCOLS + col0 + l16] = val;
  }
}

extern "C" void kernel_launch(void* const* d_in, const int* in_sizes, int n_in,
                              void* d_out, int out_size, void* d_ws, size_t ws_size,
                              hipStream_t stream) {
  const float* x   = (const float*)d_in[0];
  const int*   ei  = (const int*)  d_in[1];
  const float* W1l = (const float*)d_in[2];
  const float* W1r = (const float*)d_in[3];
  const float* b1  = (const float*)d_in[4];
  const float* W2l = (const float*)d_in[5];
  const float* W2r = (const float*)d_in[6];
  const float* b2  = (const float*)d_in[7];
  float* out = (float*)d_out;

  // workspace layout (floats): deg[N] | agg[N*128] | h[N*128]
  float* deg = (float*)d_ws;
  float* agg = deg + N_NODES;
  float* h   = agg + (size_t)N_NODES * 128;

  const int* src = ei;             // edge_index[0]
  const int* dst = ei + N_EDGES;   // edge_index[1]

  const long zn1 = (long)N_NODES * (1 + 128);      // deg + agg (contiguous)
  const long zn2 = (long)N_NODES * 128;            // agg only
  const long scat_threads = (long)N_EDGES * 32;

  dim3 blk256(256);
  int zgrid1 = (int)((zn1 + 255) / 256);
  int zgrid2 = (int)((zn2 + 255) / 256);
  int dgrid  = (N_EDGES + 255) / 256;
  int sgrid  = (int)((scat_threads + 255) / 256);
  int ggrid  = N_NODES / 16;  // 6250, exact

  // ---- layer 1 ----
  sage_zero_f32<<<zgrid1, blk256, 0, stream>>>(deg, zn1);
  sage_degree  <<<dgrid,  blk256, 0, stream>>>(dst, deg, N_EDGES);
  sage_scatter <<<sgrid,  blk256, 0, stream>>>(x, src, dst, agg, N_EDGES);
  sage_gemm<HID_C, true><<<ggrid, dim3((HID_C / 16) * 32), 0, stream>>>(
      agg, deg, x, W1l, W1r, b1, h);

  // ---- layer 2 (degree unchanged; re-zero agg, scatter h) ----
  sage_zero_f32<<<zgrid2, blk256, 0, stream>>>(agg, zn2);
  sage_scatter <<<sgrid,  blk256, 0, stream>>>(h, src, dst, agg, N_EDGES);
  sage_gemm<OUT_C, false><<<ggrid, dim3((OUT_C / 16) * 32), 0, stream>>>(
      agg, deg, h, W2l, W2r, b2, out);
}